// QuantumIntegrator_781684048358
// MI455X (gfx1250) — compile-verified
//
#include <hip/hip_runtime.h>
#include <cmath>

typedef __attribute__((ext_vector_type(2))) float v2f;
typedef __attribute__((ext_vector_type(4))) float v4f;
typedef __attribute__((ext_vector_type(8))) float v8f;

#define S_DIM 4096
#define ROWS_PER_BLOCK 128
#define NTHREADS 256
#define EPSF 1e-6f
#define RADIUS_UPDATE_RATE 0.01f
#define MIN_RADIUS 0.1f
#define MAX_RADIUS 2.0f
#define FLT_MAX_F 3.402823466e+38f

// finite guard without branches: exponent != all-ones
__device__ __forceinline__ float finite_or(float v, float fb) {
    return ((__float_as_uint(v) & 0x7f800000u) != 0x7f800000u) ? v : fb;
}

// single-instruction relu: v_med3_f32(x, 0, FLT_MAX) == clamp(x, 0, FLT_MAX)
__device__ __forceinline__ float relu1(float x) {
    return __builtin_amdgcn_fmed3f(x, 0.0f, FLT_MAX_F);
}

__device__ __forceinline__ v8f wmma_rank2(v2f a, v2f b) {
    v8f c = {};
    // D = A(16x4) * B(4x16) + 0 ; K=2..3 are zero padding
    return __builtin_amdgcn_wmma_f32_16x16x4_f32(
        /*neg_a=*/false, a, /*neg_b=*/false, b,
        /*c_mod=*/(short)0, c, /*reuse_a=*/false, /*reuse_b=*/false);
}

// ---------------------------------------------------------------------------
// Phase 1: score (normalized, causal) -> d_out; k1, r_mid, ssum -> workspace
// ---------------------------------------------------------------------------
__global__ __launch_bounds__(NTHREADS)
void qint_phase1(const float* __restrict__ r_embed, const float* __restrict__ dt,
                 float* __restrict__ score_out, float* __restrict__ r_mid_ws,
                 float* __restrict__ k1_ws, float* __restrict__ ssum_ws)
{
    __shared__ v2f s_r[S_DIM];                       // 32 KB: whole batch of r
    const int blocks_per_batch = S_DIM / ROWS_PER_BLOCK;   // 32
    const int b     = blockIdx.x / blocks_per_batch;
    const int chunk = blockIdx.x % blocks_per_batch;
    const int tid   = threadIdx.x;
    const int lane  = tid & 31;
    const int hi    = lane >> 4;                     // 0: lanes 0-15, 1: lanes 16-31
    const int nl    = lane & 15;

    const v2f* rb = (const v2f*)(r_embed + (size_t)b * S_DIM * 2);
    for (int i = tid; i < S_DIM; i += NTHREADS) s_r[i] = rb[i];
    __syncthreads();

    // wave-uniform row-strip base (readfirstlane => scalar loop control)
    const int RM = __builtin_amdgcn_readfirstlane(chunk * ROWS_PER_BLOCK + (tid >> 5) * 16);

    // A fragment (loop invariant): lanes 0-15 carry K=0,1 = (x,y); upper half zero pad
    v2f arow = s_r[RM + nl];
    v2f afrag; afrag.x = hi ? 0.0f : arow.x; afrag.y = hi ? 0.0f : arow.y;

    // causal mask for the diagonal tile, as a 0/1 multiplier
    float dmask[8];
    #pragma unroll
    for (int j = 0; j < 8; ++j) dmask[j] = (nl <= j + 8 * hi) ? 1.0f : 0.0f;

    float acc_s[8], acc_x[8], acc_y[8];
    #pragma unroll
    for (int j = 0; j < 8; ++j) { acc_s[j] = 0.f; acc_x[j] = 0.f; acc_y[j] = 0.f; }

    // -------- pass 1: full tiles below the diagonal (branch/mask free) --------
    for (int CT = 0; CT < RM; CT += 16) {
        v2f rc = s_r[CT + nl];
        v2f bfrag; bfrag.x = hi ? 0.0f : rc.x; bfrag.y = hi ? 0.0f : rc.y;
        v8f d = wmma_rank2(afrag, bfrag);
        #pragma unroll
        for (int j = 0; j < 8; ++j) {
            float v = relu1(d[j]);
            acc_s[j] += v;
            acc_x[j] = fmaf(v, rc.x, acc_x[j]);
            acc_y[j] = fmaf(v, rc.y, acc_y[j]);
        }
    }
    // peeled diagonal tile (CT == RM): relu + causal mask multiply
    {
        v2f rc = s_r[RM + nl];
        v2f bfrag; bfrag.x = hi ? 0.0f : rc.x; bfrag.y = hi ? 0.0f : rc.y;
        v8f d = wmma_rank2(afrag, bfrag);
        #pragma unroll
        for (int j = 0; j < 8; ++j) {
            float v = relu1(d[j]) * dmask[j];
            acc_s[j] += v;
            acc_x[j] = fmaf(v, rc.x, acc_x[j]);
            acc_y[j] = fmaf(v, rc.y, acc_y[j]);
        }
    }

    // cross-lane reduce over N (halves stay separate: rows j / j+8)
    #pragma unroll
    for (int j = 0; j < 8; ++j) {
        #pragma unroll
        for (int off = 1; off < 16; off <<= 1) {
            acc_s[j] += __shfl_xor(acc_s[j], off, 32);
            acc_x[j] += __shfl_xor(acc_x[j], off, 32);
            acc_y[j] += __shfl_xor(acc_y[j], off, 32);
        }
    }

    const float dtb = dt[b];
    float invd[8];
    #pragma unroll
    for (int j = 0; j < 8; ++j) invd[j] = 1.0f / fmaxf(acc_s[j], EPSF);

    // per-row results: lane nl==j of each half writes row RM + j + 8*hi
    #pragma unroll
    for (int j = 0; j < 8; ++j) {
        if (nl == j) {
            const int row = RM + j + 8 * hi;
            const float cx = acc_x[j] * invd[j];
            const float cy = acc_y[j] * invd[j];
            const float k1x = finite_or(dtb * cx, 0.0f);
            const float k1y = finite_or(dtb * cy, 0.0f);
            const v2f rv = s_r[row];
            const float mx = finite_or(rv.x + k1x, rv.x);
            const float my = finite_or(rv.y + k1y, rv.y);
            const size_t o2 = ((size_t)b * S_DIM + row) * 2;
            k1_ws[o2] = k1x;  k1_ws[o2 + 1] = k1y;
            r_mid_ws[o2] = mx; r_mid_ws[o2 + 1] = my;
            ssum_ws[(size_t)b * S_DIM + row] = acc_s[j] * invd[j];
        }
    }

    // -------- pass 2: recompute tiles, normalize, store score --------
    float* sbase = score_out
        + ((size_t)b * S_DIM + (size_t)(RM + 8 * hi)) * (size_t)S_DIM + nl;
    for (int CT = 0; CT < RM; CT += 16) {
        v2f rc = s_r[CT + nl];
        v2f bfrag; bfrag.x = hi ? 0.0f : rc.x; bfrag.y = hi ? 0.0f : rc.y;
        v8f d = wmma_rank2(afrag, bfrag);
        #pragma unroll
        for (int j = 0; j < 8; ++j)
            sbase[(size_t)j * S_DIM + CT] = relu1(d[j]) * invd[j];
    }
    // peeled diagonal tile
    {
        v2f rc = s_r[RM + nl];
        v2f bfrag; bfrag.x = hi ? 0.0f : rc.x; bfrag.y = hi ? 0.0f : rc.y;
        v8f d = wmma_rank2(afrag, bfrag);
        #pragma unroll
        for (int j = 0; j < 8; ++j)
            sbase[(size_t)j * S_DIM + RM] = relu1(d[j]) * dmask[j] * invd[j];
    }

    // -------- zero fill above the diagonal with b128 stores --------
    const int z0 = RM + 16;
    const int span = S_DIM - z0;                 // multiple of 16
    const int nchunk = span >> 6;                // 64-float chunks per half-wave row
    const int rem = span & 63;                   // 0,16,32,48
    const v4f zero4 = {};
    #pragma unroll
    for (int j = 0; j < 8; ++j) {
        float* rowp = score_out
            + ((size_t)b * S_DIM + (size_t)(RM + j + 8 * hi)) * (size_t)S_DIM;
        int col = z0 + nl * 4;
        for (int it = 0; it < nchunk; ++it, col += 64)
            *(v4f*)(rowp + col) = zero4;         // 16 lanes x float4 per row
        if (nl * 4 < rem)
            *(v4f*)(rowp + col) = zero4;         // 16/32/48-float tail
    }
}

// ---------------------------------------------------------------------------
// Phase 2: same sweep on r_mid (no stores) -> k2 -> Heun step + radius clamp
// ---------------------------------------------------------------------------
__global__ __launch_bounds__(NTHREADS)
void qint_phase2(const float* __restrict__ r_embed, const float* __restrict__ dt,
                 const float* __restrict__ r_mid_ws, const float* __restrict__ k1_ws,
                 const float* __restrict__ ssum_ws, float* __restrict__ r_final_out)
{
    __shared__ v2f s_m[S_DIM];
    const int blocks_per_batch = S_DIM / ROWS_PER_BLOCK;
    const int b     = blockIdx.x / blocks_per_batch;
    const int chunk = blockIdx.x % blocks_per_batch;
    const int tid   = threadIdx.x;
    const int lane  = tid & 31;
    const int hi    = lane >> 4;
    const int nl    = lane & 15;

    const v2f* mb = (const v2f*)(r_mid_ws + (size_t)b * S_DIM * 2);
    for (int i = tid; i < S_DIM; i += NTHREADS) s_m[i] = mb[i];
    __syncthreads();

    const int RM = __builtin_amdgcn_readfirstlane(chunk * ROWS_PER_BLOCK + (tid >> 5) * 16);

    v2f arow = s_m[RM + nl];
    v2f afrag; afrag.x = hi ? 0.0f : arow.x; afrag.y = hi ? 0.0f : arow.y;

    float dmask[8];
    #pragma unroll
    for (int j = 0; j < 8; ++j) dmask[j] = (nl <= j + 8 * hi) ? 1.0f : 0.0f;

    float acc_s[8], acc_x[8], acc_y[8];
    #pragma unroll
    for (int j = 0; j < 8; ++j) { acc_s[j] = 0.f; acc_x[j] = 0.f; acc_y[j] = 0.f; }

    for (int CT = 0; CT < RM; CT += 16) {
        v2f rc = s_m[CT + nl];
        v2f bfrag; bfrag.x = hi ? 0.0f : rc.x; bfrag.y = hi ? 0.0f : rc.y;
        v8f d = wmma_rank2(afrag, bfrag);
        #pragma unroll
        for (int j = 0; j < 8; ++j) {
            float v = relu1(d[j]);
            acc_s[j] += v;
            acc_x[j] = fmaf(v, rc.x, acc_x[j]);
            acc_y[j] = fmaf(v, rc.y, acc_y[j]);
        }
    }
    {
        v2f rc = s_m[RM + nl];
        v2f bfrag; bfrag.x = hi ? 0.0f : rc.x; bfrag.y = hi ? 0.0f : rc.y;
        v8f d = wmma_rank2(afrag, bfrag);
        #pragma unroll
        for (int j = 0; j < 8; ++j) {
            float v = relu1(d[j]) * dmask[j];
            acc_s[j] += v;
            acc_x[j] = fmaf(v, rc.x, acc_x[j]);
            acc_y[j] = fmaf(v, rc.y, acc_y[j]);
        }
    }

    #pragma unroll
    for (int j = 0; j < 8; ++j) {
        #pragma unroll
        for (int off = 1; off < 16; off <<= 1) {
            acc_s[j] += __shfl_xor(acc_s[j], off, 32);
            acc_x[j] += __shfl_xor(acc_x[j], off, 32);
            acc_y[j] += __shfl_xor(acc_y[j], off, 32);
        }
    }

    const float dtb = dt[b];
    #pragma unroll
    for (int j = 0; j < 8; ++j) {
        if (nl == j) {
            const int row = RM + j + 8 * hi;
            const float invd = 1.0f / fmaxf(acc_s[j], EPSF);
            const float k2x = finite_or(dtb * (acc_x[j] * invd), 0.0f);
            const float k2y = finite_or(dtb * (acc_y[j] * invd), 0.0f);
            const size_t o2 = ((size_t)b * S_DIM + row) * 2;
            const float rx = r_embed[o2], ry = r_embed[o2 + 1];
            const float k1x = k1_ws[o2],  k1y = k1_ws[o2 + 1];
            const float rnx = rx + 0.5f * (k1x + k2x);
            const float rny = ry + 0.5f * (k1y + k2y);
            float nr = sqrtf(rnx * rnx + rny * rny);
            nr = fmaxf(nr, EPSF);
            const float adj = ssum_ws[(size_t)b * S_DIM + row] * RADIUS_UPDATE_RATE;
            const float ar = fminf(fmaxf(nr + adj, MIN_RADIUS), MAX_RADIUS);
            const float scale = ar / nr;
            r_final_out[o2]     = finite_or(rnx * scale, rx);
            r_final_out[o2 + 1] = finite_or(rny * scale, ry);
        }
    }
}

extern "C" void kernel_launch(void* const* d_in, const int* in_sizes, int n_in,
                              void* d_out, int out_size, void* d_ws, size_t ws_size,
                              hipStream_t stream) {
    const float* r_embed = (const float*)d_in[0];
    const float* dt      = (const float*)d_in[1];
    const int B = in_sizes[1];                       // dt has one entry per batch
    float* out     = (float*)d_out;
    float* r_final = out;                            // [B, S, 2]
    float* score   = out + (size_t)B * S_DIM * 2;    // [B, S, S]

    float* ws    = (float*)d_ws;
    float* r_mid = ws;                               // B*S*2 floats
    float* k1    = ws + (size_t)B * S_DIM * 2;       // B*S*2 floats
    float* ssum  = ws + (size_t)2 * B * S_DIM * 2;   // B*S floats

    const int grid = B * (S_DIM / ROWS_PER_BLOCK);
    qint_phase1<<<grid, NTHREADS, 0, stream>>>(r_embed, dt, score, r_mid, k1, ssum);
    qint_phase2<<<grid, NTHREADS, 0, stream>>>(r_embed, dt, r_mid, k1, ssum, r_final);
}